// GATConv_76012331205023
// MI455X (gfx1250) — compile-verified
//
#include <hip/hip_runtime.h>

#define N_NODES 100000
#define N_EDGES 1280000
#define CH 64
#define LDW 66   // transposed-W LDS stride: even (8B-aligned b64 loads), bank stride 2

typedef float v2f __attribute__((ext_vector_type(2)));
typedef float v8f __attribute__((ext_vector_type(8)));

// ---------------- kernel 1: zero scratch (degrees) ----------------
__global__ void zero_f32_kernel(float* __restrict__ p, int n) {
    int i = blockIdx.x * 256 + threadIdx.x;
    if (i < n) p[i] = 0.0f;
}

// ---------------- kernel 2: out = broadcast(bias) ----------------
__global__ void init_out_kernel(float* __restrict__ out, const float* __restrict__ b) {
    int i = blockIdx.x * 256 + threadIdx.x;
    if (i < N_NODES * CH) out[i] = b[i & (CH - 1)];
}

// ---------------- kernel 3: degree accumulation ----------------
__global__ void degree_kernel(const long long* __restrict__ src,
                              const long long* __restrict__ dst,
                              float* __restrict__ outdeg,
                              float* __restrict__ indeg) {
    int e = blockIdx.x * 256 + threadIdx.x;
    if (e < N_EDGES) {
        unsafeAtomicAdd(&outdeg[(int)src[e]], 1.0f);
        unsafeAtomicAdd(&indeg[(int)dst[e]], 1.0f);
    }
}

// ---------------- kernel 4: deg -> rsqrt(deg) (0 if deg==0), in place ----------------
__global__ void norm_kernel(float* __restrict__ deg, int n) {
    int i = blockIdx.x * 256 + threadIdx.x;
    if (i < n) {
        float v = deg[i];
        deg[i] = (v > 0.0f) ? rsqrtf(v) : 0.0f;
    }
}

// ---------------- kernel 5: h = (x * norm_src) @ W  via V_WMMA_F32_16X16X4_F32 ----------------
// Block = 256 threads = 8 wave32; each wave owns a 16-row strip, computes 16x64 output.
// A (16x4 f32): lane L holds row = L&15, K-pair base kb = (L>>4)*2, values (kb, kb+1) in a v2f.
// B (4x16 f32): lane L holds col N = L&15, same K striping -> needs {W[kb][col], W[kb+1][col]}.
//   W is stored TRANSPOSED in LDS (Wt[col*LDW + row]) so the pair is one contiguous,
//   8B-aligned ds_load_b64 per fragment (no marshaling movs, conflict-free banks).
// C/D (16x16 f32): VGPR r: lanes 0-15 -> M=r, lanes 16-31 -> M=r+8; N = lane&15.
__global__ void __launch_bounds__(256) gemm_wmma_kernel(const float* __restrict__ x,
                                                        const float* __restrict__ W,
                                                        const float* __restrict__ norm_src,
                                                        float* __restrict__ h) {
    __shared__ float Wt[CH * LDW];   // 16.5 KB, W transposed + padded
    const int t = threadIdx.x;
    {   // cooperative transposing load: 1024 float4s / 256 threads = 4 each
        const float4* Ws = (const float4*)W;
#pragma unroll
        for (int i = 0; i < 4; ++i) {
            int lin = t + i * 256;        // float4 index within W
            int row = lin >> 4;           // 0..63 (16 float4 per row)
            int c4  = (lin & 15) * 4;     // starting col of this float4
            float4 f = Ws[lin];
            Wt[(c4 + 0) * LDW + row] = f.x;
            Wt[(c4 + 1) * LDW + row] = f.y;
            Wt[(c4 + 2) * LDW + row] = f.z;
            Wt[(c4 + 3) * LDW + row] = f.w;
        }
    }
    __syncthreads();

    const int lane = t & 31;
    const int wave = t >> 5;                      // wave32
    const int m0 = (blockIdx.x * 8 + wave) * 16;  // 16-row strip per wave
    if (m0 >= N_NODES) return;                    // wave-uniform (N_NODES % 16 == 0)

    const int mrow = m0 + (lane & 15);
    const int kh = lane >> 4;                     // 0 or 1: which K-pair half
    const float ns = norm_src[mrow];

    // Preload & pre-scale all 16 A fragments (K = 64 -> 16 steps of 4)
    v2f a[16];
    const float* xr = x + (size_t)mrow * CH + kh * 2;
#pragma unroll
    for (int kk = 0; kk < 16; ++kk) {
        v2f av = *(const v2f*)(xr + kk * 4);
        a[kk].x = av.x * ns;
        a[kk].y = av.y * ns;
    }

    const int nc = lane & 15;
#pragma unroll
    for (int nt = 0; nt < 4; ++nt) {              // 4 column tiles of 16
        v8f acc = {0.f, 0.f, 0.f, 0.f, 0.f, 0.f, 0.f, 0.f};
        const int col = nt * 16 + nc;
        const float* bcol = &Wt[col * LDW + kh * 2];
#pragma unroll
        for (int kk = 0; kk < 16; ++kk) {
            v2f bv = *(const v2f*)(bcol + kk * 4);  // single ds_load_b64
            acc = __builtin_amdgcn_wmma_f32_16x16x4_f32(
                /*neg_a=*/false, a[kk], /*neg_b=*/false, bv,
                /*c_mod=*/(short)0, acc, /*reuse_a=*/false, /*reuse_b=*/false);
        }
        float* hp = h + (size_t)(m0 + kh * 8) * CH + col;
#pragma unroll
        for (int r = 0; r < 8; ++r) hp[(size_t)r * CH] = acc[r];
    }
}

// ---------------- kernel 6: scatter-add  out[dst] += h[src] * norm_dst[dst] ----------------
// 16 threads per edge, float4 per thread -> 4 global_atomic_add_f32 each.
__global__ void scatter_kernel(const float* __restrict__ h,
                               const long long* __restrict__ src,
                               const long long* __restrict__ dst,
                               const float* __restrict__ norm_dst,
                               float* __restrict__ out) {
    long long tid = (long long)blockIdx.x * 256 + threadIdx.x;
    int e = (int)(tid >> 4);
    int c = (int)(tid & 15) * 4;
    if (e < N_EDGES) {
        int s = (int)src[e];
        int d = (int)dst[e];
        float nd = norm_dst[d];
        float4 v = *(const float4*)(h + (size_t)s * CH + c);
        float* o = out + (size_t)d * CH + c;
        unsafeAtomicAdd(o + 0, v.x * nd);
        unsafeAtomicAdd(o + 1, v.y * nd);
        unsafeAtomicAdd(o + 2, v.z * nd);
        unsafeAtomicAdd(o + 3, v.w * nd);
    }
}

extern "C" void kernel_launch(void* const* d_in, const int* in_sizes, int n_in,
                              void* d_out, int out_size, void* d_ws, size_t ws_size,
                              hipStream_t stream) {
    const float*     x   = (const float*)d_in[0];
    const long long* ei  = (const long long*)d_in[1];   // int64 per reference
    const float*     W   = (const float*)d_in[2];
    const float*     b   = (const float*)d_in[3];
    float* out = (float*)d_out;

    // workspace layout (floats): [outdeg N][indeg N][h N*64]
    float* ws     = (float*)d_ws;
    float* outdeg = ws;
    float* indeg  = ws + N_NODES;
    float* h      = ws + 2 * N_NODES;

    const long long* src = ei;
    const long long* dst = ei + N_EDGES;

    zero_f32_kernel<<<(2 * N_NODES + 255) / 256, 256, 0, stream>>>(outdeg, 2 * N_NODES);
    init_out_kernel<<<(N_NODES * CH + 255) / 256, 256, 0, stream>>>(out, b);
    degree_kernel<<<(N_EDGES + 255) / 256, 256, 0, stream>>>(src, dst, outdeg, indeg);
    norm_kernel<<<(2 * N_NODES + 255) / 256, 256, 0, stream>>>(outdeg, 2 * N_NODES);
    gemm_wmma_kernel<<<(N_NODES + 127) / 128, 256, 0, stream>>>(x, W, outdeg /*norm_src*/, h);
    scatter_kernel<<<(N_EDGES * 16) / 256, 256, 0, stream>>>(h, src, dst, indeg /*norm_dst*/, out);
}